// STICKYKVCache_LayerWise_14637248545497
// MI455X (gfx1250) — compile-verified
//
#include <hip/hip_runtime.h>
#include <stddef.h>

// ---------------- problem constants (match reference module constants) ------
#define NH          16
#define SEQ         4096
#define HD          128
#define MAXCTX      8192
#define MAXWIN      128
#define SINK        5
#define OMEGA       64
#define KWIN        3
#define LOCALTOK    256
#define NUMWIN      58          // (max(SINK, SEQ-LOCALTOK-OMEGA) - SINK) / OMEGA
#define REVIEW_END  3717        // SINK + NUMWIN*OMEGA
#define RECENT0     3776        // SEQ - LOCALTOK - OMEGA
#define NRECENT     320         // LOCALTOK + OMEGA
#define KP          517         // SINK + KWIN*OMEGA + NRECENT

// ---------------- flat output offsets (floats), reference return order ------
#define OFF_K       0                         // (1,16,517,128)
#define OFF_V       1058816                   // (1,16,517,128)
#define OFF_WIN     2117632                   // (16,58)
#define OFF_IMP     2118560                   // (16,4096)
#define OFF_WS      2184096                   // (16,128,3)
#define OFF_LEDGER  2190240                   // (8192,34)
#define OFF_HIST    2468768                   // (8192,16)

typedef __attribute__((ext_vector_type(2))) float v2f;
typedef __attribute__((ext_vector_type(8))) float v8f;

// ---------------------------------------------------------------------------
// Kernel 1: zero the importance accumulator (it is accumulated via atomics).
__global__ void zero_imp_kernel(float* __restrict__ imp) {
    int idx = blockIdx.x * blockDim.x + threadIdx.x;
    if (idx < NH * SEQ) imp[idx] = 0.0f;
}

// ---------------------------------------------------------------------------
// Kernel 2: importance_map[h][j] = sum_i attn[h][i][j], via ones-A WMMA.
// Each wave owns a 16-column stripe; B tile = 4 rows x 16 cols per WMMA.
// B layout (4x16 f32): lane&15 selects column N; wave halves carry K rows.
// With A == ones, C[m][n] accumulates the column sums of all streamed tiles.
__global__ void colsum_wmma_kernel(const float* __restrict__ attn,
                                   float* __restrict__ imp) {
    const int h     = blockIdx.y;          // head
    const int ctile = blockIdx.x;          // 0..31 : 128 columns per block
    const int chunk = blockIdx.z;          // 0..7  : 512 rows per chunk
    const int lane  = threadIdx.x & 31;
    const int wave  = threadIdx.x >> 5;    // 0..7  : 16 columns per wave

    const int col   = ctile * 128 + wave * 16 + (lane & 15);
    const int rhalf = (lane >> 4) * 2;     // lanes 0-15 -> rows r,r+1 ; 16-31 -> r+2,r+3

    const float* p = attn + ((size_t)h * SEQ + (size_t)chunk * 512 + rhalf) * SEQ + col;

    v2f a; a.x = 1.0f; a.y = 1.0f;         // ones A-matrix (16x4)
    v8f acc = {};                          // zero C

    #pragma unroll 4
    for (int r = 0; r < 512; r += 4) {
        v2f b;
        b.x = p[r * SEQ];                  // row r + rhalf
        b.y = p[r * SEQ + SEQ];            // row r + rhalf + 1
        acc = __builtin_amdgcn_wmma_f32_16x16x4_f32(
            /*neg_a=*/false, a, /*neg_b=*/false, b,
            /*c_mod=*/(short)0, acc, /*reuse_a=*/false, /*reuse_b=*/false);
    }
    // Every row of C equals the column sums; lane<16 holds N=lane (M=0 row).
    if (lane < 16) atomicAdd(&imp[h * SEQ + col], acc[0]);
}

// ---------------------------------------------------------------------------
// Kernel 3: per head -> window sums, top-3 window selection (sorted ascending
// into d_ws), win_scores output and full ws (16,128,3) output.
__global__ void window_topk_kernel(const float* __restrict__ imp,
                                   const float* __restrict__ ws_in,
                                   float* __restrict__ out_win,
                                   float* __restrict__ out_ws,
                                   int* __restrict__ wsort) {
    const int h = blockIdx.x;
    const int t = threadIdx.x;             // 64 threads
    __shared__ float win[NUMWIN];

    if (t < NUMWIN) {
        const float* p = imp + h * SEQ + SINK + t * OMEGA;
        float s = 0.0f;
        #pragma unroll
        for (int q = 0; q < OMEGA; ++q) s += p[q];
        win[t] = s;
        out_win[h * NUMWIN + t] = s;
    }
    __syncthreads();

    // copy the (128,3) slab for this head from the input window_scores
    for (int idx = t; idx < MAXWIN * 3; idx += 64)
        out_ws[h * MAXWIN * 3 + idx] = ws_in[h * MAXWIN * 3 + idx];
    __syncthreads();

    if (t < NUMWIN) {
        out_ws[h * MAXWIN * 3 + t * 3 + 0] = win[t];
        out_ws[h * MAXWIN * 3 + t * 3 + 1] = (float)t;
        out_ws[h * MAXWIN * 3 + t * 3 + 2] = (float)t;
    }

    if (t == 0) {
        unsigned long long used = 0ull;
        int picks[KWIN];
        for (int s = 0; s < KWIN; ++s) {
            float best = -3.4e38f; int bi = 0;
            for (int w = 0; w < NUMWIN; ++w)
                if (!((used >> w) & 1ull) && win[w] > best) { best = win[w]; bi = w; }
            used |= (1ull << bi);
            picks[s] = bi;
        }
        int a = picks[0], b = picks[1], c = picks[2], tm;
        if (a > b) { tm = a; a = b; b = tm; }
        if (b > c) { tm = b; b = c; c = tm; }
        if (a > b) { tm = a; a = b; b = tm; }
        wsort[h * 3 + 0] = a; wsort[h * 3 + 1] = b; wsort[h * 3 + 2] = c;
    }
}

// ---------------------------------------------------------------------------
// Kernel 4: gather survivors into updated_k / updated_v (float4 copies).
__global__ void gather_kv_kernel(const float* __restrict__ k,
                                 const float* __restrict__ v,
                                 const int* __restrict__ wsort,
                                 float* __restrict__ out_k,
                                 float* __restrict__ out_v) {
    const int p = blockIdx.x;              // 0..516 physical slot
    const int h = blockIdx.y;
    const int t = threadIdx.x;             // 32 threads, float4 each

    int tok;
    if (p < SINK)            tok = p;
    else if (p < SINK + KWIN * OMEGA) {
        int r = (p - SINK) >> 6, q = (p - SINK) & 63;
        tok = SINK + wsort[h * 3 + r] * OMEGA + q;
    } else                   tok = RECENT0 + (p - (SINK + KWIN * OMEGA));

    const float4* ks = (const float4*)(k + ((size_t)h * SEQ + tok) * HD);
    const float4* vs = (const float4*)(v + ((size_t)h * SEQ + tok) * HD);
    float4* kd = (float4*)(out_k + ((size_t)h * KP + p) * HD);
    float4* vd = (float4*)(out_v + ((size_t)h * KP + p) * HD);
    kd[t] = ks[t];
    vd[t] = vs[t];
}

// ---------------------------------------------------------------------------
// Kernel 5: ledger (8192,34) and hist (8192,16). Survivor physical slot is
// closed-form from the 3 sorted window indices.
__global__ void ledger_hist_kernel(const float* __restrict__ ledger_in,
                                   const float* __restrict__ hist_in,
                                   const float* __restrict__ imp,
                                   const int* __restrict__ wsort,
                                   float* __restrict__ out_ledger,
                                   float* __restrict__ out_hist) {
    const int i = blockIdx.x;              // token row 0..8191
    const int t = threadIdx.x;             // 64 threads

    if (t < 2 + 2 * NH) {                  // ledger columns
        float val;
        if (t == 0)      val = (i < SEQ) ? (float)i : ledger_in[i * 34 + 0];
        else if (t == 1) val = (i < SEQ) ? 0.0f     : ledger_in[i * 34 + 1];
        else if (t < 2 + NH) {             // physical-slot column per head
            int h = t - 2;
            val = -1.0f;
            if (i < SINK)                       val = (float)i;
            else if (i >= RECENT0 && i < SEQ)   val = (float)(SINK + KWIN * OMEGA + i - RECENT0);
            else if (i < REVIEW_END) {
                int j = (i - SINK) >> 6, q = (i - SINK) & 63;
                if      (j == wsort[h * 3 + 0]) val = (float)(SINK + q);
                else if (j == wsort[h * 3 + 1]) val = (float)(SINK + OMEGA + q);
                else if (j == wsort[h * 3 + 2]) val = (float)(SINK + 2 * OMEGA + q);
            }
        } else {                           // importance column per head
            int h = t - (2 + NH);
            val = (i < SEQ) ? imp[h * SEQ + i] : ledger_in[i * 34 + t];
        }
        out_ledger[i * 34 + t] = val;
    } else if (t < 2 + 2 * NH + NH) {      // hist columns
        int h = t - (2 + 2 * NH);
        out_hist[i * NH + h] = (i < SEQ) ? imp[h * SEQ + i] : hist_in[i * NH + h];
    }
}

// ---------------------------------------------------------------------------
extern "C" void kernel_launch(void* const* d_in, const int* in_sizes, int n_in,
                              void* d_out, int out_size, void* d_ws, size_t ws_size,
                              hipStream_t stream) {
    const float* attn      = (const float*)d_in[0];
    const float* k         = (const float*)d_in[1];
    const float* v         = (const float*)d_in[2];
    const float* ledger_in = (const float*)d_in[3];
    const float* ws_in     = (const float*)d_in[4];
    const float* hist_in   = (const float*)d_in[5];

    float* out   = (float*)d_out;
    int*   wsort = (int*)d_ws;             // 16 heads x 3 sorted window ids

    // 1) zero importance accumulator
    zero_imp_kernel<<<(NH * SEQ + 255) / 256, 256, 0, stream>>>(out + OFF_IMP);

    // 2) bandwidth-bound column-sum over 1 GiB via WMMA ones-matmul
    dim3 gridR(32, NH, 8);                 // 32 col-tiles x 16 heads x 8 row-chunks
    colsum_wmma_kernel<<<gridR, 256, 0, stream>>>(attn, out + OFF_IMP);

    // 3) window sums + top-3 + ws output
    window_topk_kernel<<<NH, 64, 0, stream>>>(out + OFF_IMP, ws_in,
                                              out + OFF_WIN, out + OFF_WS, wsort);

    // 4) survivor K/V gather
    dim3 gridG(KP, NH);
    gather_kv_kernel<<<gridG, 32, 0, stream>>>(k, v, wsort,
                                               out + OFF_K, out + OFF_V);

    // 5) ledger + history
    ledger_hist_kernel<<<MAXCTX, 64, 0, stream>>>(ledger_in, hist_in,
                                                  out + OFF_IMP, wsort,
                                                  out + OFF_LEDGER, out + OFF_HIST);
}